// GAT_10943576670615
// MI455X (gfx1250) — compile-verified
//
#include <hip/hip_runtime.h>
#include <hip/hip_bf16.h>
#include <math.h>
#include <stdint.h>

// ---- problem constants (match reference) ----
#define INDIM     128
#define HEADS     4
#define CH        64
#define FDIM      256      // HEADS * CH
#define OUTDIM    10
#define NEG_SLOPE 0.2f
#define BN_EPS    1e-5f

typedef __attribute__((ext_vector_type(2))) float v2f;
typedef __attribute__((ext_vector_type(8))) float v8f;
typedef int v4i_ __attribute__((vector_size(16)));   // pointee type expected by
                                                     // global_load_async_to_lds_b128

#define AS1 __attribute__((address_space(1)))
#define AS3 __attribute__((address_space(3)))

#if defined(__has_builtin)
# if __has_builtin(__builtin_amdgcn_global_load_async_to_lds_b128) && \
     __has_builtin(__builtin_amdgcn_s_wait_asynccnt)
#  define USE_ASYNC_LDS 1
# endif
#endif
#ifndef USE_ASYNC_LDS
# define USE_ASYNC_LDS 0
#endif

// ============================================================================
// GEMM: C[M,Ncol] = A[M,K] @ B[K,Ncol], fp32 via V_WMMA_F32_16X16X4_F32.
// Block = 256 thr (8 waves). Block owns a 128-row x 64-col output patch:
//   - whole block async-stages the B[K x 64] panel into LDS (<= 64 KB),
//   - each wave computes one 16-row x 64-col strip with 4 WMMA accumulators.
// Per-lane operand layout (ISA 7.12.2, 32-bit):
//   A/B: lane l, vgpr j  -> K = 2*(l>>4)+j, M/N = l&15
//   C/D: lane l, vgpr v  -> M = v + 8*(l>>4), N = l&15
// EXEC stays all-1s around every WMMA (wave-uniform control flow only).
// ============================================================================
__global__ void __launch_bounds__(256)
gemm_wmma_f32(const float* __restrict__ A, const float* __restrict__ B,
              float* __restrict__ C, int M, int K, int Ncol) {
  __shared__ float Bs[256 * 64];               // K(<=256) x 64 panel, 64 KB
  const int tid  = threadIdx.x;
  const int lane = tid & 31;
  const int wave = tid >> 5;
  const int tj0  = blockIdx.y << 6;            // 64-column strip base

  // ---- cooperative global -> LDS stage of the B panel (ASYNCcnt path) ----
  const int nVec = K << 4;                     // (K*64)/4 float4 elements
  for (int idx = tid; idx < nVec; idx += 256) {  // uniform trip count
    const int k  = idx >> 4;
    const int c4 = (idx & 15) << 2;
    const float* gsrc = B + (size_t)k * Ncol + tj0 + c4;
    float* ldst = &Bs[(k << 6) + c4];
#if USE_ASYNC_LDS
    __builtin_amdgcn_global_load_async_to_lds_b128(
        (AS1 v4i_*)(uintptr_t)gsrc, (AS3 v4i_*)(uintptr_t)ldst, 0, 0);
#else
    *(float4*)ldst = *(const float4*)gsrc;
#endif
  }
#if USE_ASYNC_LDS
  __builtin_amdgcn_s_wait_asynccnt(0);
#endif
  __syncthreads();                             // all threads reach this

  const int ti = blockIdx.x * 8 + wave;        // wave-uniform row tile
  if (ti >= (M >> 4)) return;                  // whole-wave exit after barrier

  const int lo = lane & 15;
  const int hi = lane >> 4;
  const float* __restrict__ arow = A + (size_t)((ti << 4) + lo) * K;
  v8f acc0 = {}, acc1 = {}, acc2 = {}, acc3 = {};
  for (int kk = 0; kk < K; kk += 4) {
    const int ka = kk + (hi << 1);
    const v2f a = *(const v2f*)(arow + ka);    // 8B-aligned (K mult of 4)
    const float* __restrict__ b0 = &Bs[(ka << 6) + lo];
    v2f b;
    b.x = b0[0];       b.y = b0[64];
    acc0 = __builtin_amdgcn_wmma_f32_16x16x4_f32(false, a, false, b, (short)0, acc0, false, false);
    b.x = b0[16];      b.y = b0[64 + 16];
    acc1 = __builtin_amdgcn_wmma_f32_16x16x4_f32(false, a, false, b, (short)0, acc1, false, false);
    b.x = b0[32];      b.y = b0[64 + 32];
    acc2 = __builtin_amdgcn_wmma_f32_16x16x4_f32(false, a, false, b, (short)0, acc2, false, false);
    b.x = b0[48];      b.y = b0[64 + 48];
    acc3 = __builtin_amdgcn_wmma_f32_16x16x4_f32(false, a, false, b, (short)0, acc3, false, false);
  }
  float* __restrict__ crow =
      C + (size_t)((ti << 4) + (hi << 3)) * Ncol + tj0 + lo;
#pragma unroll
  for (int v = 0; v < 8; ++v) {
    float* r = crow + (size_t)v * Ncol;
    r[0] = acc0[v]; r[16] = acc1[v]; r[32] = acc2[v]; r[48] = acc3[v];
  }
}

// ============================================================================
// s[n,h] = <xw[n,h,:], a_src[h,:]> ; d[n,h] = <xw[n,h,:], a_dst[h,:]>
// ============================================================================
__global__ void attn_scores(const float* __restrict__ xw,
                            const float* __restrict__ a_src,
                            const float* __restrict__ a_dst,
                            float* __restrict__ s, float* __restrict__ d, int n) {
  const int idx = blockIdx.x * blockDim.x + threadIdx.x;
  if (idx >= n * HEADS) return;
  const int node = idx >> 2, h = idx & 3;
  const float* __restrict__ xr = xw + (size_t)node * FDIM + h * CH;
  const float* __restrict__ as = a_src + h * CH;
  const float* __restrict__ ad = a_dst + h * CH;
  float ss = 0.f, dd = 0.f;
#pragma unroll 8
  for (int c = 0; c < CH; ++c) { const float v = xr[c]; ss += v * as[c]; dd += v * ad[c]; }
  s[idx] = ss; d[idx] = dd;
}

// ---- small helpers ----
__global__ void fill_f32(float* __restrict__ p, float val, long n) {
  const long i = (long)blockIdx.x * blockDim.x + threadIdx.x;
  if (i < n) p[i] = val;
}
__global__ void init_bias(float* __restrict__ h, const float* __restrict__ b, int n) {
  const long i = (long)blockIdx.x * blockDim.x + threadIdx.x;
  if (i < (long)n * FDIM) h[i] = b[i & (FDIM - 1)];
}
__global__ void relu_inplace(float* __restrict__ h, long n) {
  const long i = (long)blockIdx.x * blockDim.x + threadIdx.x;
  if (i < n) h[i] = fmaxf(h[i], 0.f);
}

// float atomic max via monotone int/uint mapping (init must be -inf)
__device__ __forceinline__ void atomicMaxF(float* addr, float val) {
  if (val >= 0.f) atomicMax((int*)addr, __float_as_int(val));
  else            atomicMin((unsigned int*)addr, __float_as_uint(val));
}

__device__ __forceinline__ float leaky(float a) {
  return a > 0.f ? a : NEG_SLOPE * a;
}

__device__ __forceinline__ void edge_endpoints(const int* __restrict__ eidx,
                                               int e, int E, int& src, int& dst) {
  if (e < E) { src = eidx[e]; dst = eidx[E + e]; }
  else       { src = dst = e - E; }             // self loops appended
}

// ============================================================================
// Pass A: m[dst,h] = max over incoming edges of leaky(s[src,h]+d[dst,h])
// ============================================================================
__global__ void edge_max(const float* __restrict__ s, const float* __restrict__ d,
                         const int* __restrict__ eidx, int E, int E2,
                         float* __restrict__ m) {
  const int e = blockIdx.x * blockDim.x + threadIdx.x;
  if (e >= E2) return;
  int src, dst; edge_endpoints(eidx, e, E, src, dst);
  const float4 sv = *(const float4*)(s + (size_t)src * 4);
  const float4 dv = *(const float4*)(d + (size_t)dst * 4);
  float* mp = m + (size_t)dst * 4;
  atomicMaxF(mp + 0, leaky(sv.x + dv.x));
  atomicMaxF(mp + 1, leaky(sv.y + dv.y));
  atomicMaxF(mp + 2, leaky(sv.z + dv.z));
  atomicMaxF(mp + 3, leaky(sv.w + dv.w));
}

// ============================================================================
// Pass B: denom[dst,h] += exp(alpha - m[dst,h])
// ============================================================================
__global__ void edge_denom(const float* __restrict__ s, const float* __restrict__ d,
                           const float* __restrict__ m,
                           const int* __restrict__ eidx, int E, int E2,
                           float* __restrict__ denom) {
  const int e = blockIdx.x * blockDim.x + threadIdx.x;
  if (e >= E2) return;
  int src, dst; edge_endpoints(eidx, e, E, src, dst);
  const float4 sv = *(const float4*)(s + (size_t)src * 4);
  const float4 dv = *(const float4*)(d + (size_t)dst * 4);
  const float4 mv = *(const float4*)(m + (size_t)dst * 4);
  float* dp = denom + (size_t)dst * 4;
  atomicAdd(dp + 0, expf(leaky(sv.x + dv.x) - mv.x));
  atomicAdd(dp + 1, expf(leaky(sv.y + dv.y) - mv.y));
  atomicAdd(dp + 2, expf(leaky(sv.z + dv.z) - mv.z));
  atomicAdd(dp + 3, expf(leaky(sv.w + dv.w) - mv.w));
}

// ============================================================================
// Pass C: out[dst,:] += w[h] * xw[src,:]  (one wave per edge, lanes over F)
// xw (51.2 MB) and out (51.2 MB) both fit in the 192 MB L2 -> atomics hit L2.
// ============================================================================
__global__ void __launch_bounds__(256)
edge_aggregate(const float* __restrict__ xw,
               const float* __restrict__ s, const float* __restrict__ d,
               const float* __restrict__ m, const float* __restrict__ denom,
               const int* __restrict__ eidx, int E, int E2,
               float* __restrict__ out) {
  const int e = blockIdx.x * 8 + (threadIdx.x >> 5);  // wave-uniform
  if (e >= E2) return;
  const int lane = threadIdx.x & 31;
  int src, dst; edge_endpoints(eidx, e, E, src, dst);
  const float4 sv = *(const float4*)(s + (size_t)src * 4);
  const float4 dv = *(const float4*)(d + (size_t)dst * 4);
  const float4 mv = *(const float4*)(m + (size_t)dst * 4);
  const float4 nv = *(const float4*)(denom + (size_t)dst * 4);
  float w[HEADS];
  w[0] = expf(leaky(sv.x + dv.x) - mv.x) / (nv.x + 1e-16f);
  w[1] = expf(leaky(sv.y + dv.y) - mv.y) / (nv.y + 1e-16f);
  w[2] = expf(leaky(sv.z + dv.z) - mv.z) / (nv.z + 1e-16f);
  w[3] = expf(leaky(sv.w + dv.w) - mv.w) / (nv.w + 1e-16f);
  const float* __restrict__ xr = xw + (size_t)src * FDIM;
  float* __restrict__ orow = out + (size_t)dst * FDIM;
#pragma unroll
  for (int it = 0; it < FDIM / 32; ++it) {
    const int c = lane + (it << 5);
    atomicAdd(orow + c, w[c >> 6] * xr[c]);
  }
}

// ============================================================================
// BatchNorm: coalesced partial sums (thread == column), then finalize + apply.
// ============================================================================
__global__ void __launch_bounds__(256)
bn_partial(const float* __restrict__ h, int n,
           float* __restrict__ sum, float* __restrict__ sumsq) {
  const int col = threadIdx.x;                        // 256 columns
  const int rowsPer = (n + gridDim.x - 1) / gridDim.x;
  const int r0 = blockIdx.x * rowsPer;
  const int r1 = min(n, r0 + rowsPer);
  float s0 = 0.f, s1 = 0.f;
  for (int r = r0; r < r1; ++r) {
    const float v = h[(size_t)r * FDIM + col];
    s0 += v; s1 += v * v;
  }
  atomicAdd(sum + col, s0);
  atomicAdd(sumsq + col, s1);
}
__global__ void bn_finalize(float* __restrict__ sum, float* __restrict__ sumsq, int n) {
  const int c = threadIdx.x;                          // one block of 256
  const float mu = sum[c] / (float)n;
  const float var = sumsq[c] / (float)n - mu * mu;    // biased (ddof=0) = jnp.var
  sum[c] = mu;
  sumsq[c] = rsqrtf(var + BN_EPS);
}
__global__ void bn_apply(float* __restrict__ h,
                         const float* __restrict__ mu, const float* __restrict__ rstd,
                         const float* __restrict__ g, const float* __restrict__ be,
                         long total) {
  const long i = (long)blockIdx.x * blockDim.x + threadIdx.x;
  if (i >= total) return;
  const int c = (int)(i & (FDIM - 1));
  h[i] = (h[i] - mu[c]) * rstd[c] * g[c] + be[c];
}

// ============================================================================
// Classifier: out[n,o] = h[n,:] @ Wc[:,o] + bc[o]   (OUT=10, tiny)
// ============================================================================
__global__ void classifier(const float* __restrict__ h, const float* __restrict__ Wc,
                           const float* __restrict__ bc, float* __restrict__ out, int n) {
  const int idx = blockIdx.x * blockDim.x + threadIdx.x;
  if (idx >= n * OUTDIM) return;
  const int node = idx / OUTDIM, o = idx - node * OUTDIM;
  const float* __restrict__ hr = h + (size_t)node * FDIM;
  float acc = bc[o];
#pragma unroll 8
  for (int c = 0; c < FDIM; ++c) acc += hr[c] * Wc[c * OUTDIM + o];
  out[idx] = acc;
}

// ============================================================================
// host-side orchestration
// ============================================================================
static inline int cdiv(long a, int b) { return (int)((a + b - 1) / b); }

extern "C" void kernel_launch(void* const* d_in, const int* in_sizes, int n_in,
                              void* d_out, int out_size, void* d_ws, size_t ws_size,
                              hipStream_t stream) {
  const float* x      = (const float*)d_in[0];
  const int*   eidx   = (const int*)  d_in[1];
  const float* W1     = (const float*)d_in[2];
  const float* asrc1  = (const float*)d_in[3];
  const float* adst1  = (const float*)d_in[4];
  const float* b1     = (const float*)d_in[5];
  const float* W2     = (const float*)d_in[6];
  const float* asrc2  = (const float*)d_in[7];
  const float* adst2  = (const float*)d_in[8];
  const float* b2     = (const float*)d_in[9];
  const float* g1     = (const float*)d_in[10];
  const float* be1    = (const float*)d_in[11];
  const float* g2     = (const float*)d_in[12];
  const float* be2    = (const float*)d_in[13];
  const float* Wc     = (const float*)d_in[14];
  const float* bc     = (const float*)d_in[15];

  const int N  = in_sizes[0] / INDIM;     // 50000
  const int E  = in_sizes[1] / 2;         // 1.6M
  const int E2 = E + N;                   // + self loops

  // ---- workspace layout (all fp32, 16B-aligned slices) ----
  char* w = (char*)d_ws;
  const size_t szNF = (size_t)N * FDIM * sizeof(float);
  const size_t szNH = (size_t)N * HEADS * sizeof(float);
  float* xw    = (float*)w;                 w += szNF;
  float* h1    = (float*)w;                 w += szNF;
  float* h2    = (float*)w;                 w += szNF;
  float* sS    = (float*)w;                 w += szNH;
  float* sD    = (float*)w;                 w += szNH;
  float* sM    = (float*)w;                 w += szNH;
  float* sDen  = (float*)w;                 w += szNH;
  float* bnSum = (float*)w;                 w += FDIM * sizeof(float);
  float* bnSq  = (float*)w;                 w += FDIM * sizeof(float);

  const long nNF = (long)N * FDIM;
  const long nNH = (long)N * HEADS;
  const int  TB  = 256;

  auto gat_layer = [&](const float* hin, int Kdim, const float* Wl,
                       const float* as, const float* ad, const float* bl,
                       const float* gl, const float* bel, float* hout) {
    // 1) xw = hin @ Wl  (WMMA fp32, LDS-staged B panel)
    dim3 ggrid(cdiv(N / 16, 8), FDIM / 64);
    gemm_wmma_f32<<<ggrid, TB, 0, stream>>>(hin, Wl, xw, N, Kdim, FDIM);
    // 2) attention logits per (node, head)
    attn_scores<<<cdiv(nNH, TB), TB, 0, stream>>>(xw, as, ad, sS, sD, N);
    // 3) scatter softmax (max, denom)
    fill_f32<<<cdiv(nNH, TB), TB, 0, stream>>>(sM, -INFINITY, nNH);
    fill_f32<<<cdiv(nNH, TB), TB, 0, stream>>>(sDen, 0.f, nNH);
    edge_max  <<<cdiv(E2, TB), TB, 0, stream>>>(sS, sD, eidx, E, E2, sM);
    edge_denom<<<cdiv(E2, TB), TB, 0, stream>>>(sS, sD, sM, eidx, E, E2, sDen);
    // 4) weighted aggregation into bias-initialized output
    init_bias<<<cdiv(nNF, TB), TB, 0, stream>>>(hout, bl, N);
    edge_aggregate<<<cdiv(E2, 8), TB, 0, stream>>>(xw, sS, sD, sM, sDen,
                                                   eidx, E, E2, hout);
    // 5) relu then batchnorm (training-mode stats)
    relu_inplace<<<cdiv(nNF, TB), TB, 0, stream>>>(hout, nNF);
    fill_f32<<<1, TB, 0, stream>>>(bnSum, 0.f, FDIM);
    fill_f32<<<1, TB, 0, stream>>>(bnSq,  0.f, FDIM);
    bn_partial <<<512, TB, 0, stream>>>(hout, N, bnSum, bnSq);
    bn_finalize<<<1, TB, 0, stream>>>(bnSum, bnSq, N);
    bn_apply<<<cdiv(nNF, TB), TB, 0, stream>>>(hout, bnSum, bnSq, gl, bel, nNF);
  };

  gat_layer(x,  INDIM, W1, asrc1, adst1, b1, g1, be1, h1);
  gat_layer(h1, FDIM,  W2, asrc2, adst2, b2, g2, be2, h2);

  classifier<<<cdiv((long)N * OUTDIM, TB), TB, 0, stream>>>(h2, Wc, bc,
                                                            (float*)d_out, N);
}